// UniGATConv_81020263071816
// MI455X (gfx1250) — compile-verified
//
#include <hip/hip_runtime.h>
#include <stdint.h>

typedef __attribute__((ext_vector_type(2))) float v2f;
typedef __attribute__((ext_vector_type(8))) float v8f;

#define CH 128          // IN_CH == H*C == 128
#define NHEAD 4
#define NUM_E 500000    // hyperedge count (constant in reference)

// ---------------------------------------------------------------------------
// X0 = X @ W via V_WMMA_F32_16X16X4_F32 (fp32 matrix core path).
// Block: 256 threads = 8 waves. Block tile: 16 rows x 128 cols.
// Wave w computes the 16x16 tile at columns [16w, 16w+16).
// ---------------------------------------------------------------------------
__global__ __launch_bounds__(256) void k_gemm_wmma(
    const float* __restrict__ X, const float* __restrict__ W,
    float* __restrict__ X0, int n) {
  __shared__ float sX[16 * 132];  // stride 132 -> conflict-free column reads
  const int t = threadIdx.x;
  const int row0 = blockIdx.x * 16;

  // Cooperative, coalesced load of the 16x128 X tile (512 float4).
  for (int i = t; i < 512; i += 256) {
    int r  = i >> 5;         // 32 float4 per row
    int c4 = i & 31;
    int rg = row0 + r; if (rg >= n) rg = n - 1;
    float4 v = ((const float4*)(X + (size_t)rg * CH))[c4];
    *((float4*)(&sX[r * 132 + c4 * 4])) = v;
  }
  __syncthreads();

  const int wave  = t >> 5;
  const int lane  = t & 31;
  const int col0  = wave * 16;
  const int m     = lane & 15;           // row (A) / col (B,C,D) index
  const int khalf = (lane >> 4) << 1;    // lanes 0-15: K=0,1  lanes 16-31: K=2,3

  v8f acc = {};
  for (int k = 0; k < CH; k += 4) {
    const int kk = k + khalf;
    v2f a, b;
    // A frag (16x4): lane holds X[row0+m][kk], X[row0+m][kk+1]
    a.x = sX[m * 132 + kk];
    a.y = sX[m * 132 + kk + 1];
    // B frag (4x16): lane holds W[kk][col0+m], W[kk+1][col0+m] (L2-resident)
    b.x = W[(size_t)kk       * CH + col0 + m];
    b.y = W[(size_t)(kk + 1) * CH + col0 + m];
    acc = __builtin_amdgcn_wmma_f32_16x16x4_f32(
        /*neg_a=*/false, a, /*neg_b=*/false, b,
        /*c_mod=*/(short)0, acc, /*reuse_a=*/false, /*reuse_b=*/false);
  }

  // C/D layout: VGPR r, lanes 0-15 -> M=r; lanes 16-31 -> M=r+8.
  const int rbase = (lane < 16) ? 0 : 8;
#pragma unroll
  for (int r = 0; r < 8; ++r) {
    int row = row0 + rbase + r;
    if (row < n) X0[(size_t)row * CH + col0 + m] = acc[r];
  }
}

// ---------------------------------------------------------------------------
// Workspace init (graph replays must be deterministic, so re-init each call).
// ---------------------------------------------------------------------------
__global__ __launch_bounds__(256) void k_init(
    float* __restrict__ esum, float* __restrict__ cnt,
    float* __restrict__ mmax, float* __restrict__ den,
    long long nes, long long ne, long long nm) {
  long long i = (long long)blockIdx.x * 256 + threadIdx.x;
  if (i < nes) esum[i] = 0.0f;
  if (i < ne)  cnt[i]  = 0.0f;
  if (i < nm)  { mmax[i] = -3.0e38f; den[i] = 0.0f; }
}

// ---------------------------------------------------------------------------
// esum[e,:] += X0[vertex[i],:]  ;  cnt[e] += 1    (one thread per nnz*channel)
// ---------------------------------------------------------------------------
__global__ __launch_bounds__(256) void k_scatter_edge(
    const float* __restrict__ X0, const int* __restrict__ vertex,
    const int* __restrict__ edges, float* __restrict__ esum,
    float* __restrict__ cnt, long long nnz) {
  long long idx = (long long)blockIdx.x * 256 + threadIdx.x;
  if (idx >= nnz * CH) return;
  long long i = idx >> 7;
  int c = (int)(idx & 127);
  size_t v = (size_t)vertex[i];
  size_t e = (size_t)edges[i];
  atomicAdd(&esum[e * CH + c], X0[v * CH + c]);
  if (c == 0) atomicAdd(&cnt[e], 1.0f);
}

// ---------------------------------------------------------------------------
// alpha_e[e,h] = sum_c( esum[e,h,c]/max(cnt,1) * att[h,c] )
// 128 threads/block = 4 wave32's = 4 heads; shuffle reduction per wave.
// ---------------------------------------------------------------------------
__global__ __launch_bounds__(128) void k_edge_alpha(
    const float* __restrict__ esum, const float* __restrict__ cnt,
    const float* __restrict__ att, float* __restrict__ alpha_e, int nedges) {
  int e = blockIdx.x;
  if (e >= nedges) return;
  int c = threadIdx.x;  // 0..127 == h*32 + (c&31)
  float inv = 1.0f / fmaxf(cnt[e], 1.0f);
  float p = esum[(size_t)e * CH + c] * inv * att[c];
#pragma unroll
  for (int off = 16; off > 0; off >>= 1) p += __shfl_xor(p, off, 32);
  if ((c & 31) == 0) alpha_e[(size_t)e * NHEAD + (c >> 5)] = p;
}

__device__ __forceinline__ float leaky01(float x) {
  return x > 0.0f ? x : 0.01f * x;
}

// float atomic-max via int max (val>=0) / uint min (val<0); correct for the
// -3e38 init and mixed-sign traffic, lowers to global_atomic_{max,min}.
__device__ __forceinline__ void atomicMaxF(float* addr, float val) {
  if (val >= 0.0f) atomicMax((int*)addr, __float_as_int(val));
  else             atomicMin((unsigned int*)addr, __float_as_uint(val));
}

// ---------------------------------------------------------------------------
// Segment max over vertex groups of leaky_relu(alpha_e[edges[i],h]).
// ---------------------------------------------------------------------------
__global__ __launch_bounds__(256) void k_seg_max(
    const int* __restrict__ vertex, const int* __restrict__ edges,
    const float* __restrict__ alpha_e, float* __restrict__ mmax, long long nnz) {
  long long idx = (long long)blockIdx.x * 256 + threadIdx.x;
  if (idx >= nnz * NHEAD) return;
  long long i = idx >> 2;
  int h = (int)(idx & 3);
  float a = leaky01(alpha_e[(size_t)edges[i] * NHEAD + h]);
  atomicMaxF(&mmax[(size_t)vertex[i] * NHEAD + h], a);
}

// ---------------------------------------------------------------------------
// ex = exp(alpha - m[v]) ; den[v] += ex ; cache ex per (nnz,h).
// ---------------------------------------------------------------------------
__global__ __launch_bounds__(256) void k_seg_exp(
    const int* __restrict__ vertex, const int* __restrict__ edges,
    const float* __restrict__ alpha_e, const float* __restrict__ mmax,
    float* __restrict__ exbuf, float* __restrict__ den, long long nnz) {
  long long idx = (long long)blockIdx.x * 256 + threadIdx.x;
  if (idx >= nnz * NHEAD) return;
  long long i = idx >> 2;
  int h = (int)(idx & 3);
  size_t v = (size_t)vertex[i];
  float a  = leaky01(alpha_e[(size_t)edges[i] * NHEAD + h]);
  float ex = __expf(a - mmax[v * NHEAD + h]);
  exbuf[idx] = ex;
  atomicAdd(&den[v * NHEAD + h], ex);
}

// ---------------------------------------------------------------------------
// out[v,:] += (esum[e,:]/max(cnt,1)) * ex/(den[v]+eps)   (out holds X0 term)
// ---------------------------------------------------------------------------
__global__ __launch_bounds__(256) void k_final(
    const int* __restrict__ vertex, const int* __restrict__ edges,
    const float* __restrict__ esum, const float* __restrict__ cnt,
    const float* __restrict__ exbuf, const float* __restrict__ den,
    float* __restrict__ out, long long nnz) {
  long long idx = (long long)blockIdx.x * 256 + threadIdx.x;
  if (idx >= nnz * CH) return;
  long long i = idx >> 7;
  int c = (int)(idx & 127);
  int h = c >> 5;
  size_t v = (size_t)vertex[i];
  size_t e = (size_t)edges[i];
  float xe = esum[e * CH + c] / fmaxf(cnt[e], 1.0f);
  float w  = exbuf[i * NHEAD + h] / (den[v * NHEAD + h] + 1e-16f);
  atomicAdd(&out[v * CH + c], xe * w);
}

extern "C" void kernel_launch(void* const* d_in, const int* in_sizes, int n_in,
                              void* d_out, int out_size, void* d_ws, size_t ws_size,
                              hipStream_t stream) {
  const float* X      = (const float*)d_in[0];   // [N,128]
  const float* W      = (const float*)d_in[1];   // [128,128]
  const float* att    = (const float*)d_in[2];   // [1,4,32] -> flat 128
  const int*   vertex = (const int*)d_in[3];     // [NNZ]
  const int*   edges  = (const int*)d_in[4];     // [NNZ]

  const int       n      = in_sizes[0] / CH;
  const long long nnz    = in_sizes[3];
  const long long nedges = NUM_E;

  // Workspace carve-up (floats): ~295 MB total.
  float* esum    = (float*)d_ws;                         // nedges*128
  float* cnt     = esum    + (size_t)nedges * CH;        // nedges
  float* alpha_e = cnt     + (size_t)nedges;             // nedges*4
  float* mmax    = alpha_e + (size_t)nedges * NHEAD;     // n*4
  float* den     = mmax    + (size_t)n * NHEAD;          // n*4
  float* exbuf   = den     + (size_t)n * NHEAD;          // nnz*4

  float* out = (float*)d_out;

  // 1) X0 = X @ W  (WMMA fp32) -> d_out (fully overwrites it)
  k_gemm_wmma<<<(n + 15) / 16, 256, 0, stream>>>(X, W, out, n);

  // 2) init workspace
  const long long nes = nedges * CH;
  const long long nm  = (long long)n * NHEAD;
  k_init<<<(int)((nes + 255) / 256), 256, 0, stream>>>(esum, cnt, mmax, den,
                                                       nes, nedges, nm);

  // 3) scatter-sum X0 rows into hyperedges
  const long long tot_c = nnz * CH;
  k_scatter_edge<<<(int)((tot_c + 255) / 256), 256, 0, stream>>>(
      out, vertex, edges, esum, cnt, nnz);

  // 4) per-edge attention logits
  k_edge_alpha<<<(int)nedges, 128, 0, stream>>>(esum, cnt, att, alpha_e,
                                                (int)nedges);

  // 5) segment softmax over vertex groups
  const long long tot_h = nnz * NHEAD;
  k_seg_max<<<(int)((tot_h + 255) / 256), 256, 0, stream>>>(
      vertex, edges, alpha_e, mmax, nnz);
  k_seg_exp<<<(int)((tot_h + 255) / 256), 256, 0, stream>>>(
      vertex, edges, alpha_e, mmax, exbuf, den, nnz);

  // 6) weighted scatter back to nodes (accumulates onto X0 already in d_out)
  k_final<<<(int)((tot_c + 255) / 256), 256, 0, stream>>>(
      vertex, edges, esum, cnt, exbuf, den, out, nnz);
}